// Transition_8186207666610
// MI455X (gfx1250) — compile-verified
//
#include <hip/hip_runtime.h>
#include <math.h>

typedef float v2f __attribute__((ext_vector_type(2)));
typedef float v8f __attribute__((ext_vector_type(8)));

constexpr int B = 128;    // batch rows
constexpr int H = 1024;   // hidden states
#define RB 256            // reduction block size

// ---------------------------------------------------------------------------
// Kernel 1: rowLSE[i] = logsumexp_j U[i, :]   (one block per row)
// ---------------------------------------------------------------------------
__global__ __launch_bounds__(RB) void row_lse_kernel(const float* __restrict__ U,
                                                     float* __restrict__ rowLSE) {
    const int row = blockIdx.x;
    const float* u = U + (size_t)row * H;
    __shared__ float red[RB];

    float m = -INFINITY;
    for (int j = threadIdx.x; j < H; j += RB) m = fmaxf(m, u[j]);
    red[threadIdx.x] = m;
    __syncthreads();
    for (int s = RB / 2; s > 0; s >>= 1) {
        if (threadIdx.x < s) red[threadIdx.x] = fmaxf(red[threadIdx.x], red[threadIdx.x + s]);
        __syncthreads();
    }
    m = red[0];
    __syncthreads();

    float sum = 0.0f;
    for (int j = threadIdx.x; j < H; j += RB) sum += expf(u[j] - m);
    red[threadIdx.x] = sum;
    __syncthreads();
    for (int s = RB / 2; s > 0; s >>= 1) {
        if (threadIdx.x < s) red[threadIdx.x] += red[threadIdx.x + s];
        __syncthreads();
    }
    if (threadIdx.x == 0) rowLSE[row] = m + logf(red[0]);
}

// ---------------------------------------------------------------------------
// Kernel 2: colmax[j] = max_i (U[i,j] - rowLSE[i])   (one thread per column)
// ---------------------------------------------------------------------------
__global__ __launch_bounds__(RB) void col_max_kernel(const float* __restrict__ U,
                                                     const float* __restrict__ rowLSE,
                                                     float* __restrict__ colmax) {
    const int j = blockIdx.x * RB + threadIdx.x;
    float c = -INFINITY;
#pragma unroll 4
    for (int i = 0; i < H; ++i) c = fmaxf(c, U[(size_t)i * H + j] - rowLSE[i]);
    colmax[j] = c;
}

// ---------------------------------------------------------------------------
// Kernel 3: build T' in K-pair-interleaved layout:
//   Tp2[(i/2)*(2H) + 2j + (i&1)] = exp(U[i,j] - rowLSE[i] - colmax[j])
// One thread handles a (row-pair, column) and writes one aligned v2f.
// ---------------------------------------------------------------------------
__global__ __launch_bounds__(RB) void build_T_kernel(const float* __restrict__ U,
                                                     const float* __restrict__ rowLSE,
                                                     const float* __restrict__ colmax,
                                                     float* __restrict__ Tp2) {
    const size_t idx = (size_t)blockIdx.x * RB + threadIdx.x;   // over (H/2)*H
    const int ip = (int)(idx / H);        // row pair index
    const int j  = (int)(idx % H);
    const int i0 = 2 * ip;
    const float cj = colmax[j];
    v2f e;
    e.x = expf(U[(size_t)i0 * H + j]       - rowLSE[i0]     - cj);
    e.y = expf(U[(size_t)(i0 + 1) * H + j] - rowLSE[i0 + 1] - cj);
    *(v2f*)(Tp2 + (size_t)ip * (2 * H) + 2 * j) = e;
}

// ---------------------------------------------------------------------------
// Kernel 4: rowmax[b] = max_i la[b,i];
//   At2[(i/2)*(2B) + 2b + (i&1)] = exp(la[b,i] - rowmax[b])   (one block / row)
// ---------------------------------------------------------------------------
__global__ __launch_bounds__(RB) void build_A_kernel(const float* __restrict__ la,
                                                     float* __restrict__ At2,
                                                     float* __restrict__ rowmax) {
    const int b = blockIdx.x;
    const float* a = la + (size_t)b * H;
    __shared__ float red[RB];

    float m = -INFINITY;
    for (int i = threadIdx.x; i < H; i += RB) m = fmaxf(m, a[i]);
    red[threadIdx.x] = m;
    __syncthreads();
    for (int s = RB / 2; s > 0; s >>= 1) {
        if (threadIdx.x < s) red[threadIdx.x] = fmaxf(red[threadIdx.x], red[threadIdx.x + s]);
        __syncthreads();
    }
    m = red[0];

    for (int ip = threadIdx.x; ip < H / 2; ip += RB) {
        v2f v = *(const v2f*)(a + 2 * ip);
        v2f e;
        e.x = expf(v.x - m);
        e.y = expf(v.y - m);
        *(v2f*)(At2 + (size_t)ip * (2 * B) + 2 * b) = e;
    }
    if (threadIdx.x == 0) rowmax[b] = m;
}

// ---------------------------------------------------------------------------
// Kernel 5: out[b,j] = log( (A' @ T')[b,j] ) + rowmax[b] + colmax[j]
// FP32 WMMA 16x16x4. One wave32 per 32x16 output tile (2 M-tiles share each
// B fragment). K-pair layouts make every fragment load one aligned b64.
//   A frag (16x4): lane l -> M = l&15, K = k0 + 2*(l>>4) + {0,1}
//   B frag (4x16): lane l -> N = l&15, K = k0 + 2*(l>>4) + {0,1}
//   C/D (16x16):   VGPR r -> M = r + 8*(l>>4), N = l&15
// ---------------------------------------------------------------------------
__global__ __launch_bounds__(32) void wmma_log_gemm_kernel(const float* __restrict__ At2,
                                                           const float* __restrict__ Tp2,
                                                           const float* __restrict__ rowmax,
                                                           const float* __restrict__ colmax,
                                                           float* __restrict__ out) {
    const int lane = threadIdx.x;
    const int half = lane >> 4;    // 0 or 1
    const int l15  = lane & 15;
    const int m0 = blockIdx.x * 32;    // output row tile pair (batch dim)
    const int n0 = blockIdx.y * 16;    // output col tile (state dim)
    const int n  = n0 + l15;

    // kp = k0/2 + half ; A offset = kp*(2B) + 2m ; B offset = kp*(2H) + 2n
    const float* pA = At2 + (size_t)half * (2 * B) + 2 * (m0 + l15);
    const float* pB = Tp2 + (size_t)half * (2 * H) + 2 * n;

    v8f acc0 = {};
    v8f acc1 = {};
#pragma unroll 8
    for (int kk = 0; kk < H / 4; ++kk) {           // k0 = 4*kk
        v2f a0 = *(const v2f*)(pA + (size_t)kk * (4 * B));        // rows m0..m0+15
        v2f a1 = *(const v2f*)(pA + (size_t)kk * (4 * B) + 32);   // rows m0+16..m0+31
        v2f bf = *(const v2f*)(pB + (size_t)kk * (4 * H));
        acc0 = __builtin_amdgcn_wmma_f32_16x16x4_f32(false, a0, false, bf,
                                                     (short)0, acc0, false, false);
        acc1 = __builtin_amdgcn_wmma_f32_16x16x4_f32(false, a1, false, bf,
                                                     (short)0, acc1, false, false);
    }

    const float cn = colmax[n];
#pragma unroll
    for (int r = 0; r < 8; ++r) {
        const int mr0 = m0 + r + 8 * half;
        const int mr1 = m0 + 16 + r + 8 * half;
        out[(size_t)mr0 * H + n] = logf(acc0[r]) + rowmax[mr0] + cn;
        out[(size_t)mr1 * H + n] = logf(acc1[r]) + rowmax[mr1] + cn;
    }
}

// ---------------------------------------------------------------------------
// Launch
// ---------------------------------------------------------------------------
extern "C" void kernel_launch(void* const* d_in, const int* in_sizes, int n_in,
                              void* d_out, int out_size, void* d_ws, size_t ws_size,
                              hipStream_t stream) {
    const float* la = (const float*)d_in[0];   // (B, H) log_alpha
    const float* U  = (const float*)d_in[1];   // (H, H) unnormalized_tran
    float* out = (float*)d_out;                // (B, H)

    float* ws_f    = (float*)d_ws;
    float* rowLSE  = ws_f;                          // H
    float* colmax  = ws_f + 1024;                   // H
    float* rowmax  = ws_f + 2048;                   // B
    float* At2     = ws_f + 4096;                   // H*B   (K-pair-interleaved A')
    float* Tp2     = ws_f + 4096 + (size_t)H * B;   // H*H   (K-pair-interleaved T')

    row_lse_kernel<<<H, RB, 0, stream>>>(U, rowLSE);
    col_max_kernel<<<H / RB, RB, 0, stream>>>(U, rowLSE, colmax);
    build_T_kernel<<<(int)(((size_t)(H / 2) * H) / RB), RB, 0, stream>>>(U, rowLSE, colmax, Tp2);
    build_A_kernel<<<B, RB, 0, stream>>>(la, At2, rowmax);

    dim3 grid(B / 32, H / 16);   // 4 x 64 = 256 wave-tiles of 32x16
    wmma_log_gemm_kernel<<<grid, 32, 0, stream>>>(At2, Tp2, rowmax, colmax, out);
}